// GraphVAE_76072460747123
// MI455X (gfx1250) — compile-verified
//
#include <hip/hip_runtime.h>
#include <math.h>

#define N 96
#define LAT 256
#define H (N*N)                 // 9216
#define OUT_DIM (N*(N+1)/2)     // 4656
#define MPM_ITERS 50

typedef __attribute__((ext_vector_type(2))) float v2f;
typedef __attribute__((ext_vector_type(8))) float v8f;

// ---- workspace layout (float offsets) ----
#define WS_PART   0             // 128*256 = 32768 encoder partial sums
#define WS_KL     32768         // 1
#define WS_P      32769         // 4656 sigmoid outputs
#define WS_D      37440         // 9216   (total 46656 floats = 187 KB)

// ---------------- Kernel 1: encoder partial GEMV (bandwidth-bound) -----------
// Block b: mat = b&1 (w_e1 / w_e2), chunk = b>>1 covers 144 rows of H.
// Thread c owns output column c -> consecutive lanes read consecutive floats.
__global__ void k_enc_partial(const float* __restrict__ gh,
                              const float* __restrict__ w_e1,
                              const float* __restrict__ w_e2,
                              float* __restrict__ ws) {
  const int b = blockIdx.x;          // 0..127
  const int mat = b & 1;
  const int chunk = b >> 1;          // 0..63
  const int c = threadIdx.x;         // 0..255
  const float* W = mat ? w_e2 : w_e1;
  const int h0 = chunk * 144;
  float acc = 0.f;
  #pragma unroll 4
  for (int h = h0; h < h0 + 144; ++h)
    acc += gh[h] * W[h * LAT + c];
  ws[WS_PART + b * LAT + c] = acc;
}

// ---------------- Kernel 2: reparam + KL + decoder MLP + sigmoid -------------
__global__ void k_decoder(const float* __restrict__ eps,
                          const float* __restrict__ b_e1,
                          const float* __restrict__ b_e2,
                          const float* __restrict__ w_d1,
                          const float* __restrict__ b_d1,
                          const float* __restrict__ w_d2,
                          const float* __restrict__ b_d2,
                          float* __restrict__ ws) {
  __shared__ float zsh[LAT];
  __shared__ float hsh[LAT];
  __shared__ float red[LAT];
  const int c = threadIdx.x;         // 0..255
  float zmu = b_e1[c], zl = b_e2[c];
  #pragma unroll 4
  for (int k = 0; k < 64; ++k) {
    zmu += ws[WS_PART + (2*k    ) * LAT + c];
    zl  += ws[WS_PART + (2*k + 1) * LAT + c];
  }
  float ez = expf(zl);
  zsh[c] = eps[c] * expf(0.5f * zl) + zmu;
  red[c] = 1.f + zl - zmu * zmu - ez;
  __syncthreads();
  for (int s = 128; s > 0; s >>= 1) {
    if (c < s) red[c] += red[c + s];
    __syncthreads();
  }
  if (c == 0) ws[WS_KL] = -0.5f * red[0] / (float)H;
  __syncthreads();
  // h = relu(z @ w_d1 + b_d1)
  float y = b_d1[c];
  #pragma unroll 4
  for (int k = 0; k < LAT; ++k) y += zsh[k] * w_d1[k * LAT + c];
  hsh[c] = fmaxf(y, 0.f);
  __syncthreads();
  // out = sigmoid(h @ w_d2 + b_d2)
  for (int o = c; o < OUT_DIM; o += LAT) {
    float acc = b_d2[o];
    #pragma unroll 4
    for (int k = 0; k < LAT; ++k) acc += hsh[k] * w_d2[k * OUT_DIM + o];
    ws[WS_P + o] = 1.f / (1.f + expf(-acc));
  }
}

// ---------------- Kernel 3: similarity setup + 50 MPM iters + loss -----------
// Single 32-wave workgroup. x, A, B, Mt all resident in LDS (~153 KB of the
// WGP's 320 KB) so the 50 serial iterations run entirely off ds_load;
// A@M uses V_WMMA_F32_16X16X4_F32 with both fragments as contiguous b64 LDS
// loads (M is stored transposed). C-tiles stay in registers across the norm
// reduction, so there is no xnew round-trip at all.
__global__ __launch_bounds__(1024)
void k_mpm(const float* __restrict__ adj,
           float* __restrict__ ws,
           float* __restrict__ out) {
  __shared__ float x[H];             // 36 KB  current assignment
  __shared__ float As[H];            // 36 KB  A[i][j]
  __shared__ float Bs[H];            // 36 KB  B[a][b]
  __shared__ float Mt[H];            // 36 KB  M transposed: Mt[a][j]
  __shared__ float red[1024];
  __shared__ float drs[N], rfs[N], afs[N];
  __shared__ int   colA[N], indA[N];
  __shared__ float invnorm;
  const int tid = threadIdx.x;
  const float* p  = ws + WS_P;
  float* Dm = ws + WS_D;

  auto iu = [](int i, int j) { return i * N - (i * (i - 1)) / 2 + (j - i); };

  // phase 1: diag + row-sum features (rec[i][j] = p[iu(min,max)])
  if (tid < N) {
    const int a = tid;
    drs[a] = p[iu(a, a)];
    float rf = 0.f, af = 0.f;
    for (int b = 0; b < N; ++b) {
      const int lo = b < a ? b : a, hi = b < a ? a : b;
      rf += p[iu(lo, hi)];
      af += adj[a * N + b];
    }
    rfs[a] = rf; afs[a] = af;
  }
  __syncthreads();

  // phase 2: A, B (LDS), D (global), x0
  for (int e = tid; e < H; e += 1024) {
    const int i = e / N, j = e % N;
    const float da_i = adj[i * N + i], da_j = adj[j * N + j];
    As[e] = (i == j) ? 0.f : adj[e] * da_i * da_j;
    const int lo = i < j ? i : j, hi = i < j ? j : i;
    Bs[e] = (i == j) ? 0.f : p[iu(lo, hi)] * drs[i] * drs[j];
    Dm[e] = da_i * drs[j] / (fabsf(afs[i] - rfs[j]) + 1.f);
    x[e] = 1.f / (float)N;
  }
  __threadfence();   // Dm visibility within the block
  __syncthreads();

  const int lane = tid & 31;
  const int wave = tid >> 5;
  const int m16  = lane & 15;
  const int kb   = (lane >> 4) * 2;   // K-pair select per ISA 16x4 f32 layout

  for (int it = 0; it < MPM_ITERS; ++it) {
    // (a) tropical half: Mt[a][j] = max_b x[j][b]*B[a][b]  (float4 LDS passes)
    for (int e = tid; e < H; e += 1024) {
      const int a = e / N, j = e % N;
      const float4* xr = (const float4*)&x[j * N];
      const float4* br = (const float4*)&Bs[a * N];
      float4 m4;
      {
        float4 xv = xr[0], bv = br[0];
        m4.x = xv.x * bv.x; m4.y = xv.y * bv.y;
        m4.z = xv.z * bv.z; m4.w = xv.w * bv.w;
      }
      #pragma unroll 4
      for (int q = 1; q < N / 4; ++q) {
        float4 xv = xr[q], bv = br[q];
        m4.x = fmaxf(m4.x, xv.x * bv.x); m4.y = fmaxf(m4.y, xv.y * bv.y);
        m4.z = fmaxf(m4.z, xv.z * bv.z); m4.w = fmaxf(m4.w, xv.w * bv.w);
      }
      Mt[e] = fmaxf(fmaxf(m4.x, m4.y), fmaxf(m4.z, m4.w));
    }
    __syncthreads();

    // (b) dense half: v = x.*D + A@M  via v_wmma_f32_16x16x4_f32 (LDS b64 ops)
    float ssq = 0.f;
    float vkeep[2][8];
    #pragma unroll
    for (int tt = 0; tt < 2; ++tt) {
      const int t = wave + tt * 32;          // 6x6 grid of 16x16 tiles
      if (t < 36) {
        const int i0 = (t / 6) * 16, a0 = (t % 6) * 16;
        v8f c = {};
        for (int k0 = 0; k0 < N; k0 += 4) {  // 24 K-steps of 4
          v2f av = *(const v2f*)&As[(i0 + m16) * N + k0 + kb];
          v2f bv = *(const v2f*)&Mt[(a0 + m16) * N + k0 + kb];
          c = __builtin_amdgcn_wmma_f32_16x16x4_f32(
              /*neg_a=*/false, av, /*neg_b=*/false, bv,
              /*c_mod=*/(short)0, c, /*reuse_a=*/false, /*reuse_b=*/false);
        }
        #pragma unroll
        for (int r = 0; r < 8; ++r) {        // C layout: lanes16-31 -> M+8
          const int row = i0 + r + (lane >> 4) * 8;
          const int col = a0 + m16;
          const float v = c[r] + x[row * N + col] * Dm[row * N + col];
          vkeep[tt][r] = v;
          ssq += v * v;
        }
      }
    }
    // (c) Frobenius norm
    red[tid] = ssq;
    __syncthreads();
    for (int s = 512; s > 0; s >>= 1) {
      if (tid < s) red[tid] += red[tid + s];
      __syncthreads();
    }
    if (tid == 0) invnorm = 1.f / (sqrtf(red[0]) + 1e-6f);
    __syncthreads();
    // (d) write normalized x straight from registers (all reads of old x done)
    #pragma unroll
    for (int tt = 0; tt < 2; ++tt) {
      const int t = wave + tt * 32;
      if (t < 36) {
        const int i0 = (t / 6) * 16, a0 = (t % 6) * 16;
        #pragma unroll
        for (int r = 0; r < 8; ++r) {
          const int row = i0 + r + (lane >> 4) * 8;
          const int col = a0 + m16;
          x[row * N + col] = vkeep[tt][r] * invnorm;
        }
      }
    }
    __syncthreads();
  }

  // phase 4: greedy row-argmax assignment (first-max, matches jnp.argmax)
  if (tid < N) {
    float best = x[tid * N]; int bi = 0;
    for (int a = 1; a < N; ++a) {
      const float v = x[tid * N + a];
      if (v > best) { best = v; bi = a; }
    }
    colA[tid] = bi;
  }
  __syncthreads();
  if (tid == 0) {
    for (int i = 0; i < N; ++i) indA[i] = 0;
    for (int i = 0; i < N; ++i) indA[colA[i]] = i;  // last-wins scatter
  }
  __syncthreads();

  // phase 5: BCE over upper triangle of permuted adjacency
  float s = 0.f;
  int k = 0;
  for (int i = 0; i < N; ++i)
    for (int j = i; j < N; ++j, ++k)
      if ((k & 1023) == tid) {
        const float tt = adj[indA[i] * N + indA[j]];
        const float pv = fminf(fmaxf(p[k], 1e-7f), 1.f - 1e-7f);
        s += tt * logf(pv) + (1.f - tt) * logf(1.f - pv);
      }
  red[tid] = s;
  __syncthreads();
  for (int sz = 512; sz > 0; sz >>= 1) {
    if (tid < sz) red[tid] += red[tid + sz];
    __syncthreads();
  }
  if (tid == 0) out[0] = -red[0] / (float)OUT_DIM + ws[WS_KL];
}

// ---------------------------------------------------------------------------
extern "C" void kernel_launch(void* const* d_in, const int* in_sizes, int n_in,
                              void* d_out, int out_size, void* d_ws, size_t ws_size,
                              hipStream_t stream) {
  const float* input_features = (const float*)d_in[0];
  const float* adj  = (const float*)d_in[1];
  const float* eps  = (const float*)d_in[2];
  const float* w_e1 = (const float*)d_in[3];
  const float* b_e1 = (const float*)d_in[4];
  const float* w_e2 = (const float*)d_in[5];
  const float* b_e2 = (const float*)d_in[6];
  const float* w_d1 = (const float*)d_in[7];
  const float* b_d1 = (const float*)d_in[8];
  const float* w_d2 = (const float*)d_in[9];
  const float* b_d2 = (const float*)d_in[10];
  float* ws  = (float*)d_ws;
  float* out = (float*)d_out;

  k_enc_partial<<<128, 256, 0, stream>>>(input_features, w_e1, w_e2, ws);
  k_decoder<<<1, 256, 0, stream>>>(eps, b_e1, b_e2, w_d1, b_d1, w_d2, b_d2, ws);
  k_mpm<<<1, 1024, 0, stream>>>(adj, ws, out);
}